// HyperbolicRegressionPoincare_18743237279917
// MI455X (gfx1250) — compile-verified
//
#include <hip/hip_runtime.h>
#include <math.h>

// ---------------------------------------------------------------------------
// Problem constants (from reference)
// ---------------------------------------------------------------------------
#define B_SZ   4096
#define IN_SZ  128
#define OUT_SZ 128

#define EPS32     1.1920928955078125e-07f   // np.finfo(float32).eps
#define CLAMP_V   16.635532333438682f       // log(2/eps) = 24*ln2
#define SMOOTH_B  50.0f
#define MIN_ENORM 1e-15f
#define MAXNORM   0.99999f                  // (1 - 1e-5)/sqrt(C), C=1

typedef __attribute__((ext_vector_type(2))) float v2f;
typedef __attribute__((ext_vector_type(8))) float v8f;

// ---------------------------------------------------------------------------
// Kernel 1: per-output-row stats and scaled A matrix
//   factor_j = clip(1 - ||bias_j||^2, EPS32)   ( = 2/lambda(bias_j) )
//   A[j][:]  = factor_j * weight[j][:]
//   uu[j]    = ||bias_j||^2
//   pa[j]    = <bias_j, A_j> = factor_j * <bias_j, weight_j>
//   an[j]    = clip(||A_j||, 1e-15) = clip(factor_j * ||weight_j||, 1e-15)
//   lp[j]    = lambda(bias_j) = 2/factor_j
// ---------------------------------------------------------------------------
__global__ void hrp_stats_kernel(const float* __restrict__ weight,
                                 const float* __restrict__ bias,
                                 float* __restrict__ Aw,
                                 float* __restrict__ uu_o,
                                 float* __restrict__ pa_o,
                                 float* __restrict__ an_o,
                                 float* __restrict__ lp_o) {
    const int j = blockIdx.x;          // 0..127
    const int t = threadIdx.x;         // 0..127
    const float bv = bias[j * IN_SZ + t];
    const float wv = weight[j * IN_SZ + t];

    __shared__ float s_bb[128];
    __shared__ float s_bw[128];
    __shared__ float s_ww[128];
    __shared__ float s_factor;

    s_bb[t] = bv * bv;
    s_bw[t] = bv * wv;
    s_ww[t] = wv * wv;
    __syncthreads();
    for (int off = 64; off > 0; off >>= 1) {
        if (t < off) {
            s_bb[t] += s_bb[t + off];
            s_bw[t] += s_bw[t + off];
            s_ww[t] += s_ww[t + off];
        }
        __syncthreads();
    }
    if (t == 0) {
        const float uu = s_bb[0];
        const float factor = fmaxf(1.0f - uu, EPS32);
        s_factor = factor;
        uu_o[j] = uu;
        pa_o[j] = factor * s_bw[0];
        an_o[j] = fmaxf(factor * sqrtf(s_ww[0]), MIN_ENORM);
        lp_o[j] = 2.0f / factor;
    }
    __syncthreads();
    Aw[j * IN_SZ + t] = s_factor * wv;
}

// ---------------------------------------------------------------------------
// Kernel 2: vv[i] = ||x_i||^2 ; one wave32 per batch row
// ---------------------------------------------------------------------------
__global__ void hrp_vv_kernel(const float* __restrict__ x,
                              float* __restrict__ vv) {
    const int lane = threadIdx.x & 31;
    const int wave = threadIdx.x >> 5;
    const int row  = blockIdx.x * (blockDim.x >> 5) + wave;
    const float* xr = x + row * IN_SZ;
    float s = 0.0f;
    #pragma unroll
    for (int k = 0; k < IN_SZ; k += 32) {
        const float v = xr[k + lane];
        s += v * v;
    }
    #pragma unroll
    for (int off = 16; off > 0; off >>= 1)
        s += __shfl_xor(s, off, 32);
    if (lane == 0) vv[row] = s;
}

// ---------------------------------------------------------------------------
// Device helpers for the epilogue
// ---------------------------------------------------------------------------
__device__ __forceinline__ float softplus_f(float z) {
    // stable: max(z,0) + log1p(exp(-|z|))
    return fmaxf(z, 0.0f) + log1pf(expf(-fabsf(z)));
}

__device__ __forceinline__ float smooth_clamp_f(float xv) {
    const float lo = -CLAMP_V, hi = CLAMP_V;
    return lo + (softplus_f(SMOOTH_B * (xv - lo)) -
                 softplus_f(SMOOTH_B * (xv - hi))) / SMOOTH_B;
}

// ---------------------------------------------------------------------------
// Kernel 3: main fused GEMM + hyperbolic epilogue.
// Block = 128 threads = 4 wave32; each wave owns a 16(M) x 16(N) tile.
// Dual f32 WMMA accumulators over K=128 in steps of 4:
//   c_uv = X * bias^T   (-> <p_j, x_i>)
//   c_bx = X * A^T      (-> <x_i, a_j>)
// A-operand layout (16x4 f32): lane<16 holds (M=lane, K=k0..k0+1),
// lane>=16 holds (M=lane-16, K=k0+2..k0+3) -> float2 load at k0+2*(lane>>4).
// B-operand (4x16): lane&15 selects N, same K split -> identical addressing
// on the row-major 128x128 bias / A matrices.
// C/D layout: VGPR r <-> (M = r + 8*(lane>>4), N = lane&15).
// ---------------------------------------------------------------------------
__global__ void __launch_bounds__(128)
hrp_main_kernel(const float* __restrict__ x,
                const float* __restrict__ bias,
                const float* __restrict__ Aw,
                const float* __restrict__ uu_a,
                const float* __restrict__ pa_a,
                const float* __restrict__ an_a,
                const float* __restrict__ lp_a,
                const float* __restrict__ vv_a,
                float* __restrict__ out) {
    const int lane = threadIdx.x & 31;
    const int wave = threadIdx.x >> 5;
    const int m0 = (blockIdx.x * 4 + wave) * 16;   // batch-tile origin
    const int n0 = blockIdx.y * 16;                // output-tile origin
    const int lm = lane & 15;
    const int kh = (lane >> 4) << 1;               // 0 or 2

    const float* xrow = x    + (size_t)(m0 + lm) * IN_SZ + kh;
    const float* prow = bias + (size_t)(n0 + lm) * IN_SZ + kh;
    const float* arow = Aw   + (size_t)(n0 + lm) * IN_SZ + kh;

    v8f c_uv = {};
    v8f c_bx = {};

    #pragma unroll
    for (int k0 = 0; k0 < IN_SZ; k0 += 4) {
        const v2f a  = *(const v2f*)(xrow + k0);
        const v2f bp = *(const v2f*)(prow + k0);
        const v2f ba = *(const v2f*)(arow + k0);
        c_uv = __builtin_amdgcn_wmma_f32_16x16x4_f32(
            false, a, false, bp, (short)0, c_uv, false, false);
        c_bx = __builtin_amdgcn_wmma_f32_16x16x4_f32(
            false, a, false, ba, (short)0, c_bx, false, false);
    }

    // ---- fused hyperbolic epilogue -------------------------------------
    const int j  = n0 + lm;
    const float uu = uu_a[j];
    const float pa = pa_a[j];
    const float an = an_a[j];
    const float lp = lp_a[j];
    const int mbase = m0 + ((lane >> 4) << 3);     // +8 for upper half-wave

    #pragma unroll
    for (int r = 0; r < 8; ++r) {
        const int i = mbase + r;
        const float vvi = vv_a[i];
        const float pxd = c_uv[r];                 // <p_j, x_i>
        const float bxd = c_bx[r];                 // <x_i, a_j>

        const float uv  = -pxd;                    // <u,v>, u = -p_j
        const float t2  = 1.0f + 2.0f * uv;
        const float den = fmaxf(t2 + uu * vvi, EPS32);
        const float alpha = (t2 + vvi) / den;      // coeff of u
        const float beta  = (1.0f - uu) / den;     // coeff of v

        float sn2  = alpha * alpha * uu + 2.0f * alpha * beta * uv
                   + beta * beta * vvi;            // ||sub||^2
        float suba = -alpha * pa + beta * bxd;     // <sub, a_j>

        // projection onto the Poincare ball
        const float nrm = fmaxf(sqrtf(sn2), MIN_ENORM);
        const float s   = (nrm > MAXNORM) ? (MAXNORM / nrm) : 1.0f;
        suba *= s;
        sn2  *= s * s;

        const float lam_sub = 2.0f / fmaxf(1.0f - sn2, EPS32);
        float arg = lam_sub * suba / an;
        arg = smooth_clamp_f(arg);
        const float sd = asinhf(arg);

        out[(size_t)i * OUT_SZ + j] = lp * an * sd;
    }
}

// ---------------------------------------------------------------------------
// Host launcher
// inputs: d_in[0]=x (4096x128 f32), d_in[1]=weight (128x128 f32),
//         d_in[2]=bias (128x128 f32); d_out: 4096x128 f32.
// workspace layout (floats): Aw[128*128] | uu[128] | pa[128] | an[128] |
//                            lp[128] | vv[4096]   (~84 KB total)
// ---------------------------------------------------------------------------
extern "C" void kernel_launch(void* const* d_in, const int* in_sizes, int n_in,
                              void* d_out, int out_size, void* d_ws, size_t ws_size,
                              hipStream_t stream) {
    const float* x      = (const float*)d_in[0];
    const float* weight = (const float*)d_in[1];
    const float* bias   = (const float*)d_in[2];
    float* out = (float*)d_out;

    float* ws = (float*)d_ws;
    float* Aw = ws;                       // 128*128
    float* uu = Aw + OUT_SZ * IN_SZ;      // 128
    float* pa = uu + OUT_SZ;              // 128
    float* an = pa + OUT_SZ;              // 128
    float* lp = an + OUT_SZ;              // 128
    float* vv = lp + OUT_SZ;              // 4096

    hrp_stats_kernel<<<dim3(OUT_SZ), dim3(IN_SZ), 0, stream>>>(
        weight, bias, Aw, uu, pa, an, lp);

    hrp_vv_kernel<<<dim3(B_SZ / 4), dim3(128), 0, stream>>>(x, vv);

    // grid: x covers M (4096 / (4 waves * 16 rows) = 64), y covers N (128/16 = 8)
    hrp_main_kernel<<<dim3(B_SZ / 64, OUT_SZ / 16), dim3(128), 0, stream>>>(
        x, bias, Aw, uu, pa, an, lp, vv, out);
}